// RebuiltGraphVAE_9509057593396
// MI455X (gfx1250) — compile-verified
//
#include <hip/hip_runtime.h>
#include <cstddef>

#define NNODE 4096
#define NFEAT 128
#define HID   256
#define NEDGE 131072

typedef float v2f __attribute__((ext_vector_type(2)));
typedef float v8f __attribute__((ext_vector_type(8)));

// ---------------------------------------------------------------------------
// Generic fp32 WMMA GEMM: C[M,N] = A[M,K] @ W[K,N] (+bias) (+relu)
// Wave32: each wave computes a 16 x (16*NT) tile with V_WMMA_F32_16X16X4_F32,
// reusing one A fragment across NT WMMAs per K-step (NT independent
// accumulation chains also hide the WMMA->WMMA RAW latency).
// Block = 256 threads = 8 waves stacked along M. grid = (N/(16*NT), M/128).
// A fragment (16x4 f32): lane L holds M=L%16, K pair (L<16 ? {k,k+1} : {k+2,k+3})
// B fragment symmetric with N in place of M. C/D: VGPR r -> M = r + (L/16)*8, N = L%16.
// ---------------------------------------------------------------------------
template <int NT>
__global__ __launch_bounds__(256)
void gemm_f32_wmma(const float* __restrict__ A, int lda,
                   const float* __restrict__ W, int ldw,
                   const float* __restrict__ bias,
                   float* __restrict__ C, int ldc,
                   int K, int relu)
{
    const int lane = threadIdx.x & 31;
    const int wave = threadIdx.x >> 5;
    const int m0 = (blockIdx.y * 8 + wave) * 16;
    const int n0 = blockIdx.x * (16 * NT);
    const int r16  = lane & 15;   // M index for A, N index for B/C
    const int half = lane >> 4;   // K half selector

    const float* __restrict__ Arow = A + (size_t)(m0 + r16) * lda;
    const float* __restrict__ Wcol = W + n0 + r16;

    v8f acc[NT];
#pragma unroll
    for (int j = 0; j < NT; ++j) acc[j] = (v8f){};

    for (int kb = 0; kb < K; kb += 4) {
        const int ka = kb + half * 2;
        v2f a;
        a[0] = Arow[ka];
        a[1] = Arow[ka + 1];
        const float* __restrict__ w0 = Wcol + (size_t)ka * ldw;
        const float* __restrict__ w1 = Wcol + (size_t)(ka + 1) * ldw;
#pragma unroll
        for (int j = 0; j < NT; ++j) {
            v2f b;
            b[0] = w0[j * 16];
            b[1] = w1[j * 16];
            acc[j] = __builtin_amdgcn_wmma_f32_16x16x4_f32(
                /*neg_a=*/false, a, /*neg_b=*/false, b,
                /*c_mod=*/(short)0, acc[j], /*reuse_a=*/false, /*reuse_b=*/false);
        }
    }

    const int mbase = m0 + half * 8;
#pragma unroll
    for (int j = 0; j < NT; ++j) {
        const int n = n0 + j * 16 + r16;
        const float bv = bias ? bias[n] : 0.0f;
#pragma unroll
        for (int r = 0; r < 8; ++r) {
            float v = acc[j][r] + bv;
            if (relu) v = fmaxf(v, 0.0f);
            C[(size_t)(mbase + r) * ldc + n] = v;
        }
    }
}

// ---------------------------------------------------------------------------
// Per-node valence head: logits = h1[i,0:32] @ w_val2[32,8] + b ; softmax ;
// argmax+1 -> pred (softmax is monotonic so argmax(logits)==argmax(valences)).
// ---------------------------------------------------------------------------
__global__ __launch_bounds__(256)
void valence_kernel(const float* __restrict__ h1,
                    const float* __restrict__ w_val2, const float* __restrict__ b_val2,
                    float* __restrict__ valences_out, float* __restrict__ pred)
{
    const int i = blockIdx.x * blockDim.x + threadIdx.x;
    if (i >= NNODE) return;
    const float* __restrict__ hrow = h1 + (size_t)i * 32;
    float lg[8];
#pragma unroll
    for (int j = 0; j < 8; ++j) lg[j] = b_val2[j];
    for (int k = 0; k < 32; ++k) {
        const float hv = hrow[k];
#pragma unroll
        for (int j = 0; j < 8; ++j) lg[j] += hv * w_val2[k * 8 + j];
    }
    float mx = lg[0]; int am = 0;
#pragma unroll
    for (int j = 1; j < 8; ++j) if (lg[j] > mx) { mx = lg[j]; am = j; }
    float ex[8], s = 0.0f;
#pragma unroll
    for (int j = 0; j < 8; ++j) { ex[j] = __expf(lg[j] - mx); s += ex[j]; }
    const float inv = 1.0f / s;
#pragma unroll
    for (int j = 0; j < 8; ++j) valences_out[(size_t)i * 8 + j] = ex[j] * inv;
    pred[i] = (float)(am + 1);
}

// ---------------------------------------------------------------------------
// Per-edge bond MLP (layer1 pre-GEMMed into B1/B2), softmax(4), bond order
// segment-sum, edge-count segment-sum, adjacency bitmask build.
// ---------------------------------------------------------------------------
__global__ __launch_bounds__(256)
void bond_kernel(const int* __restrict__ row, const int* __restrict__ col,
                 const float* __restrict__ B1, const float* __restrict__ B2,
                 const float* __restrict__ b_bond1,
                 const float* __restrict__ w_bond2, const float* __restrict__ b_bond2,
                 float* __restrict__ bond_out, float* __restrict__ ndeg,
                 float* __restrict__ degcnt, unsigned* __restrict__ bits)
{
    const int e = blockIdx.x * blockDim.x + threadIdx.x;
    if (e >= NEDGE) return;
    const int r = row[e], c = col[e];
    float lg0 = b_bond2[0], lg1 = b_bond2[1], lg2 = b_bond2[2], lg3 = b_bond2[3];
    const float* __restrict__ p1 = B1 + (size_t)r * 32;
    const float* __restrict__ p2 = B2 + (size_t)c * 32;
#pragma unroll 8
    for (int j = 0; j < 32; ++j) {
        const float hv = fmaxf(p1[j] + p2[j] + b_bond1[j], 0.0f);
        lg0 += hv * w_bond2[j * 4 + 0];
        lg1 += hv * w_bond2[j * 4 + 1];
        lg2 += hv * w_bond2[j * 4 + 2];
        lg3 += hv * w_bond2[j * 4 + 3];
    }
    const float mx = fmaxf(fmaxf(lg0, lg1), fmaxf(lg2, lg3));
    float e0 = __expf(lg0 - mx), e1 = __expf(lg1 - mx);
    float e2 = __expf(lg2 - mx), e3 = __expf(lg3 - mx);
    const float inv = 1.0f / (e0 + e1 + e2 + e3);
    e0 *= inv; e1 *= inv; e2 *= inv; e3 *= inv;
    bond_out[(size_t)e * 4 + 0] = e0;
    bond_out[(size_t)e * 4 + 1] = e1;
    bond_out[(size_t)e * 4 + 2] = e2;
    bond_out[(size_t)e * 4 + 3] = e3;
    const float order = e0 + 2.0f * e1 + 3.0f * e2 + 1.5f * e3;
    atomicAdd(&ndeg[r], order);
    atomicAdd(&degcnt[r], 1.0f);
    atomicOr(&bits[(size_t)r * 128 + (c >> 5)], 1u << (c & 31));
}

// ---------------------------------------------------------------------------
// Per-edge (one wave each): edge_tokens = P1[row] + P2[col] + b_edge, and
// nbr_sum[row] += node_tokens[col] (segment sum via f32 atomics).
// h = lane + i*32 keeps every access a fully coalesced wave-wide burst.
// ---------------------------------------------------------------------------
__global__ __launch_bounds__(256)
void edge_token_kernel(const int* __restrict__ row, const int* __restrict__ col,
                       const float* __restrict__ P1, const float* __restrict__ P2,
                       const float* __restrict__ b_edge,
                       const float* __restrict__ nt,
                       float* __restrict__ et, float* __restrict__ nbrs)
{
    const int gw = blockIdx.x * 8 + (threadIdx.x >> 5);
    const int lane = threadIdx.x & 31;
    if (gw >= NEDGE) return;
    const int r = row[gw], c = col[gw];
    const float* __restrict__ p1  = P1 + (size_t)r * HID;
    const float* __restrict__ p2  = P2 + (size_t)c * HID;
    const float* __restrict__ ntc = nt + (size_t)c * HID;
    float* __restrict__ nb = nbrs + (size_t)r * HID;
    float* __restrict__ eo = et + (size_t)gw * HID;
#pragma unroll
    for (int i2 = 0; i2 < 8; ++i2) {
        const int h = lane + i2 * 32;
        eo[h] = p1[h] + p2[h] + b_edge[h];
        atomicAdd(&nb[h], ntc[h]);
    }
}

// ---------------------------------------------------------------------------
// Valence violation: mean((ndeg - pred)^2), wave-reduced then atomicAdd.
// ---------------------------------------------------------------------------
__global__ __launch_bounds__(256)
void violation_kernel(const float* __restrict__ ndeg, const float* __restrict__ pred,
                      float* __restrict__ out)
{
    const int i = blockIdx.x * blockDim.x + threadIdx.x;
    float v = 0.0f;
    if (i < NNODE) {
        const float d = ndeg[i] - pred[i];
        v = d * d * (1.0f / (float)NNODE);
    }
    for (int off = 16; off > 0; off >>= 1) v += __shfl_down(v, off, 32);
    if ((threadIdx.x & 31) == 0) atomicAdd(out, v);
}

// ---------------------------------------------------------------------------
// 2-hop unique set via bitmasks. One 256-thread block per node:
//   mask = OR over k in N(i) of bits[k]; clear self bit;
//   expand mask -> u16 index list in LDS (shared-atomic cursor);
//   hop2_sum[h] = sum over list of node_tokens[j][h]  (h = threadIdx.x):
//   uniform ds broadcast of list[j] + one coalesced 1KB global read per j.
// Raw sums land in cat3[:,512:768]; normalized in cat_kernel.
// ---------------------------------------------------------------------------
__global__ __launch_bounds__(256)
void hop2_kernel(const unsigned* __restrict__ bits, const float* __restrict__ nt,
                 float* __restrict__ cat3, float* __restrict__ cnt2)
{
    const int i = blockIdx.x;
    const int t = threadIdx.x;
    __shared__ unsigned selfb[128];
    __shared__ unsigned mask[128];
    __shared__ unsigned short list[NNODE];
    __shared__ int lcnt;
    if (t == 0) lcnt = 0;
    if (t < 128) {
        selfb[t] = bits[(size_t)i * 128 + t];
        mask[t] = 0u;
    }
    __syncthreads();
    // Union of neighbor adjacency rows: thread t<128 owns mask word t.
    for (int w = 0; w < 128; ++w) {
        unsigned b = selfb[w];
        while (b) {
            const int bt = __builtin_ctz(b);
            b &= b - 1;
            const int k = w * 32 + bt;
            if (t < 128) mask[t] |= bits[(size_t)k * 128 + t];
        }
    }
    __syncthreads();
    if (t == 0) mask[i >> 5] &= ~(1u << (i & 31));   // exclude self
    __syncthreads();
    // Compact set bits into an index list.
    if (t < 128) {
        unsigned b = mask[t];
        const int n = __popc(b);
        if (n) {
            int pos = atomicAdd(&lcnt, n);
            const int base = t * 32;
            while (b) {
                const int bt = __builtin_ctz(b);
                b &= b - 1;
                list[pos++] = (unsigned short)(base + bt);
            }
        }
    }
    __syncthreads();
    const int cnt = lcnt;
    if (t == 0) cnt2[i] = (float)cnt;
    float s = 0.0f;
    for (int j = 0; j < cnt; ++j) {
        s += nt[(size_t)list[j] * HID + t];
    }
    cat3[(size_t)i * 768 + 512 + t] = s;
}

// ---------------------------------------------------------------------------
// Build cat3 = [node_tokens | nbr_mean | hop2_mean]  (stride 768).
// ---------------------------------------------------------------------------
__global__ __launch_bounds__(256)
void cat_kernel(const float* __restrict__ nt, const float* __restrict__ nbr_sum,
                const float* __restrict__ degcnt, const float* __restrict__ cnt2,
                float* __restrict__ cat3)
{
    const int i = blockIdx.x;
    const int h = threadIdx.x;
    const float d  = degcnt[i];
    const float c2 = cnt2[i];
    const size_t base = (size_t)i * 768;
    cat3[base + h] = nt[(size_t)i * HID + h];
    cat3[base + 256 + h] = (d > 0.0f) ? nbr_sum[(size_t)i * HID + h] / fmaxf(d, 1.0f) : 0.0f;
    const float hs = cat3[base + 512 + h];
    cat3[base + 512 + h] = (c2 > 0.0f) ? hs / fmaxf(c2, 1.0f) : 0.0f;
}

// ---------------------------------------------------------------------------
extern "C" void kernel_launch(void* const* d_in, const int* in_sizes, int n_in,
                              void* d_out, int out_size, void* d_ws, size_t ws_size,
                              hipStream_t stream)
{
    (void)in_sizes; (void)n_in; (void)out_size; (void)ws_size;

    const float* x   = (const float*)d_in[0];
    const int*   ei  = (const int*)d_in[1];
    const int*   row = ei;
    const int*   col = ei + NEDGE;
    const float* w_val1  = (const float*)d_in[2];
    const float* b_val1  = (const float*)d_in[3];
    const float* w_val2  = (const float*)d_in[4];
    const float* b_val2  = (const float*)d_in[5];
    const float* w_bond1 = (const float*)d_in[6];
    const float* b_bond1 = (const float*)d_in[7];
    const float* w_bond2 = (const float*)d_in[8];
    const float* b_bond2 = (const float*)d_in[9];
    const float* w_node  = (const float*)d_in[10];
    const float* b_node  = (const float*)d_in[11];
    const float* w_edge  = (const float*)d_in[12];
    const float* b_edge  = (const float*)d_in[13];
    const float* w_sg1   = (const float*)d_in[14];
    const float* b_sg1   = (const float*)d_in[15];
    const float* w_sg2   = (const float*)d_in[16];
    const float* b_sg2   = (const float*)d_in[17];
    const float* w_nb1   = (const float*)d_in[18];
    const float* b_nb1   = (const float*)d_in[19];
    const float* w_nb2   = (const float*)d_in[20];
    const float* b_nb2   = (const float*)d_in[21];

    // Workspace layout (floats)
    float* ws = (float*)d_ws;
    float*    h1   = ws;                              //  4096*32
    float*    B1   = ws + 131072;                     //  4096*32
    float*    B2   = ws + 262144;                     //  4096*32
    float*    P1   = ws + 393216;                     //  4096*256
    float*    P2   = ws + 1441792;                    //  4096*256
    float*    pred = ws + 2490368;                    //  4096
    float*    ndeg = ws + 2494464;                    //  4096   (zeroed region start)
    float*    degc = ws + 2498560;                    //  4096
    float*    nbrs = ws + 2502656;                    //  4096*256
    float*    cnt2 = ws + 3551232;                    //  4096
    unsigned* bits = (unsigned*)(ws + 3555328);       //  4096*128 u32 (zeroed region end)
    float*    cat3 = ws + 4079616;                    //  4096*768
    float*    sgh  = ws + 7225344;                    //  4096*256
    float*    nbh  = ws + 8273920;                    //  4096*256

    // Output layout (tuple concatenated flat, fp32)
    float* out      = (float*)d_out;
    float* out_viol = out;                 // [1]
    float* out_val  = out + 1;             // [4096,8]
    float* out_bond = out + 32769;         // [131072,4]
    float* out_nt   = out + 557057;        // [4096,256]
    float* out_et   = out + 1605633;       // [131072,256]
    float* out_sg   = out + 35160065;      // [4096,256]
    float* out_nb   = out + 36208641;      // [4096,256]

    // Zero accumulators every call (graph replays; ws is not re-poisoned).
    hipMemsetAsync(ndeg, 0, (size_t)(4096 + 4096 + 1048576 + 4096 + 524288) * 4, stream);
    hipMemsetAsync(out_viol, 0, 4, stream);

    const dim3 blk(256);

    // Valence head layer 1: h1 = relu(x @ w_val1 + b_val1)   [N=32 -> NT=2]
    gemm_f32_wmma<2><<<dim3(1, 32), blk, 0, stream>>>(x, NFEAT, w_val1, 32, b_val1, h1, 32, NFEAT, 1);
    valence_kernel<<<16, 256, 0, stream>>>(h1, w_val2, b_val2, out_val, pred);

    // Bond layer 1 factored: B1 = x @ w_bond1[:128], B2 = x @ w_bond1[128:]
    gemm_f32_wmma<2><<<dim3(1, 32), blk, 0, stream>>>(x, NFEAT, w_bond1, 32, nullptr, B1, 32, NFEAT, 0);
    gemm_f32_wmma<2><<<dim3(1, 32), blk, 0, stream>>>(x, NFEAT, w_bond1 + 128 * 32, 32, nullptr, B2, 32, NFEAT, 0);

    // node_tokens = x @ w_node + b_node  (written straight to output, reused below)
    gemm_f32_wmma<4><<<dim3(4, 32), blk, 0, stream>>>(x, NFEAT, w_node, HID, b_node, out_nt, HID, NFEAT, 0);

    // edge_tokens factored: P1 = x @ w_edge[:128], P2 = x @ w_edge[128:]
    gemm_f32_wmma<4><<<dim3(4, 32), blk, 0, stream>>>(x, NFEAT, w_edge, HID, nullptr, P1, HID, NFEAT, 0);
    gemm_f32_wmma<4><<<dim3(4, 32), blk, 0, stream>>>(x, NFEAT, w_edge + 128 * 256, HID, nullptr, P2, HID, NFEAT, 0);

    // Per-edge: bond types + bond-order segsum + degree segsum + adjacency bits
    bond_kernel<<<512, 256, 0, stream>>>(row, col, B1, B2, b_bond1, w_bond2, b_bond2,
                                         out_bond, ndeg, degc, bits);
    // Per-edge: edge tokens + neighbor-sum segsum
    edge_token_kernel<<<16384, 256, 0, stream>>>(row, col, P1, P2, b_edge, out_nt, out_et, nbrs);

    violation_kernel<<<16, 256, 0, stream>>>(ndeg, pred, out_viol);

    // 2-hop unique-set sums via adjacency bitmasks (replaces dense adj@adj)
    hop2_kernel<<<4096, 256, 0, stream>>>(bits, out_nt, cat3, cnt2);
    cat_kernel<<<4096, 256, 0, stream>>>(out_nt, nbrs, degc, cnt2, cat3);

    // subgraph_tokens = relu(cat3[:, :512] @ w_sg1 + b_sg1) @ w_sg2 + b_sg2
    gemm_f32_wmma<4><<<dim3(4, 32), blk, 0, stream>>>(cat3, 768, w_sg1, HID, b_sg1, sgh, HID, 512, 1);
    gemm_f32_wmma<4><<<dim3(4, 32), blk, 0, stream>>>(sgh, HID, w_sg2, HID, b_sg2, out_sg, HID, HID, 0);

    // neighborhood_tokens = relu(cat3 @ w_nb1 + b_nb1) @ w_nb2 + b_nb2
    gemm_f32_wmma<4><<<dim3(4, 32), blk, 0, stream>>>(cat3, 768, w_nb1, HID, b_nb1, nbh, HID, 768, 1);
    gemm_f32_wmma<4><<<dim3(4, 32), blk, 0, stream>>>(nbh, HID, w_nb2, HID, b_nb2, out_nb, HID, HID, 0);
}